// AMLModel_40776419508665
// MI455X (gfx1250) — compile-verified
//
#include <hip/hip_runtime.h>

typedef float v2f __attribute__((ext_vector_type(2)));
typedef float v8f __attribute__((ext_vector_type(8)));

#define N_NODES 100000
#define N_EDGES 1200000
#define F_IN    128
#define HIDDEN  64
#define TOT_E   (N_EDGES + N_NODES)   // edges + self-loops
#define N_WAVES (N_NODES / 16)        // 6250, exact

// ---------------- init: deg=1 (self-loop), zero accumulators ----------------
__global__ void gcn_init_kernel(float* __restrict__ deg,
                                float* __restrict__ h1,
                                float* __restrict__ o2) {
    int i = blockIdx.x * blockDim.x + threadIdx.x;
    if (i < N_NODES * HIDDEN) h1[i] = 0.0f;
    if (i < N_NODES)          deg[i] = 1.0f;   // self-loop contribution
    if (i < N_NODES * 2)      o2[i] = 0.0f;
}

// ---------------- degree accumulation over dst ----------------
__global__ void gcn_degree_kernel(const long long* __restrict__ ei,
                                  float* __restrict__ deg) {
    int e = blockIdx.x * blockDim.x + threadIdx.x;
    if (e >= N_EDGES) return;
    int d = (int)ei[(long long)N_EDGES + e];
    atomicAdd(&deg[d], 1.0f);
}

// ---------------- deg -> d^{-1/2} in place ----------------
__global__ void gcn_rsqrt_kernel(float* __restrict__ deg) {
    int i = blockIdx.x * blockDim.x + threadIdx.x;
    if (i >= N_NODES) return;
    float d = deg[i];
    deg[i] = (d > 0.0f) ? rsqrtf(d) : 0.0f;
}

// ---------------- GEMM1: H0[100000x64] = X[100000x128] @ W1[128x64] ----------
// One wave computes a 16-row x 64-col tile with 4 f32 WMMA accumulators.
__global__ void __launch_bounds__(256)
gcn_gemm1_kernel(const float* __restrict__ X,
                 const float* __restrict__ W1,
                 float* __restrict__ H0) {
    int wave = (blockIdx.x * blockDim.x + threadIdx.x) >> 5;
    if (wave >= N_WAVES) return;                 // wave-uniform: EXEC stays full
    int lane  = threadIdx.x & 31;
    int lrow  = lane & 15;                       // M (A) / N (B,C,D) within tile
    int khalf = lane >> 4;                       // 0: K={0,1}, 1: K={2,3}

    int row = wave * 16 + lrow;
    const float* xrow = X + (long)row * F_IN;

    v8f acc0 = {}, acc1 = {}, acc2 = {}, acc3 = {};

    for (int k0 = 0; k0 < F_IN; k0 += 4) {
        int ka = k0 + 2 * khalf;
        __builtin_prefetch(xrow + ka + 32, 0, 0);      // global_prefetch_b8
        v2f A;
        A.x = xrow[ka];
        A.y = xrow[ka + 1];
        const float* w0 = W1 + (long)ka * HIDDEN + lrow;        // row K=ka
        const float* w1 = W1 + (long)(ka + 1) * HIDDEN + lrow;  // row K=ka+1
        v2f B0 = { w0[0],  w1[0]  };
        v2f B1 = { w0[16], w1[16] };
        v2f B2 = { w0[32], w1[32] };
        v2f B3 = { w0[48], w1[48] };
        acc0 = __builtin_amdgcn_wmma_f32_16x16x4_f32(false, A, false, B0, (short)0, acc0, false, false);
        acc1 = __builtin_amdgcn_wmma_f32_16x16x4_f32(false, A, false, B1, (short)0, acc1, false, false);
        acc2 = __builtin_amdgcn_wmma_f32_16x16x4_f32(false, A, false, B2, (short)0, acc2, false, false);
        acc3 = __builtin_amdgcn_wmma_f32_16x16x4_f32(false, A, false, B3, (short)0, acc3, false, false);
    }

    // C/D layout: VGPR r -> M=r (lanes 0-15) or M=r+8 (lanes 16-31), N=lrow
    int mbase = wave * 16 + (khalf ? 8 : 0);
    for (int r = 0; r < 8; ++r) {
        long o = (long)(mbase + r) * HIDDEN + lrow;
        H0[o]      = acc0[r];
        H0[o + 16] = acc1[r];
        H0[o + 32] = acc2[r];
        H0[o + 48] = acc3[r];
    }
}

// ---------------- Aggregation 1: H1[dst] += norm * H0[src] (64 wide) --------
// One wave per edge; each lane moves a float2 -> coalesced gather + scatter.
__global__ void __launch_bounds__(256)
gcn_agg1_kernel(const long long* __restrict__ ei,
                const float* __restrict__ dinv,
                const float* __restrict__ H0,
                float* __restrict__ H1) {
    long gtid = (long)blockIdx.x * blockDim.x + threadIdx.x;
    long e = gtid >> 5;
    if (e >= TOT_E) return;
    int lane = threadIdx.x & 31;

    int s, d;
    if (e < N_EDGES) {
        s = (int)ei[e];
        d = (int)ei[(long long)N_EDGES + e];
    } else {
        s = d = (int)(e - N_EDGES);               // self-loop
    }
    float nrm = dinv[s] * dinv[d];

    const float2* srcp = (const float2*)(H0 + (long)s * HIDDEN);
    float2 v = srcp[lane];
    float* dstp = H1 + (long)d * HIDDEN + lane * 2;
    atomicAdd(dstp,     v.x * nrm);
    atomicAdd(dstp + 1, v.y * nrm);
}

// ---------------- GEMM2: O0[100000x2] = relu(H1 + b1) @ W2[64x2] ------------
// WMMA with B zero-padded from N=2 to N=16; bias+ReLU fused on A-load.
__global__ void __launch_bounds__(256)
gcn_gemm2_kernel(const float* __restrict__ H1,
                 const float* __restrict__ b1,
                 const float* __restrict__ W2,
                 float* __restrict__ O0) {
    int wave = (blockIdx.x * blockDim.x + threadIdx.x) >> 5;
    if (wave >= N_WAVES) return;
    int lane  = threadIdx.x & 31;
    int lrow  = lane & 15;
    int khalf = lane >> 4;

    int row = wave * 16 + lrow;
    const float* hrow = H1 + (long)row * HIDDEN;
    bool nvalid = (lrow < 2);

    v8f acc = {};
    for (int k0 = 0; k0 < HIDDEN; k0 += 4) {
        int ka = k0 + 2 * khalf;
        v2f A;
        A.x = fmaxf(hrow[ka]     + b1[ka],     0.0f);
        A.y = fmaxf(hrow[ka + 1] + b1[ka + 1], 0.0f);
        v2f B;
        B.x = nvalid ? W2[ka * 2 + lrow]       : 0.0f;
        B.y = nvalid ? W2[(ka + 1) * 2 + lrow] : 0.0f;
        acc = __builtin_amdgcn_wmma_f32_16x16x4_f32(false, A, false, B, (short)0, acc, false, false);
    }

    if (nvalid) {
        int mbase = wave * 16 + (khalf ? 8 : 0);
        for (int r = 0; r < 8; ++r)
            O0[(long)(mbase + r) * 2 + lrow] = acc[r];
    }
}

// ---------------- Aggregation 2: O2[dst] += norm * O0[src] (2 wide) ---------
__global__ void gcn_agg2_kernel(const long long* __restrict__ ei,
                                const float* __restrict__ dinv,
                                const float* __restrict__ O0,
                                float* __restrict__ O2) {
    long e = (long)blockIdx.x * blockDim.x + threadIdx.x;
    if (e >= TOT_E) return;
    int s, d;
    if (e < N_EDGES) {
        s = (int)ei[e];
        d = (int)ei[(long long)N_EDGES + e];
    } else {
        s = d = (int)(e - N_EDGES);
    }
    float nrm = dinv[s] * dinv[d];
    atomicAdd(&O2[(long)d * 2],     O0[(long)s * 2]     * nrm);
    atomicAdd(&O2[(long)d * 2 + 1], O0[(long)s * 2 + 1] * nrm);
}

// ---------------- bias + log_softmax over 2 classes ----------------
__global__ void gcn_final_kernel(const float* __restrict__ O2,
                                 const float* __restrict__ b2,
                                 float* __restrict__ out) {
    int i = blockIdx.x * blockDim.x + threadIdx.x;
    if (i >= N_NODES) return;
    float o0 = O2[(long)i * 2]     + b2[0];
    float o1 = O2[(long)i * 2 + 1] + b2[1];
    float m = fmaxf(o0, o1);
    float lse = m + logf(expf(o0 - m) + expf(o1 - m));
    out[(long)i * 2]     = o0 - lse;
    out[(long)i * 2 + 1] = o1 - lse;
}

extern "C" void kernel_launch(void* const* d_in, const int* in_sizes, int n_in,
                              void* d_out, int out_size, void* d_ws, size_t ws_size,
                              hipStream_t stream) {
    const float*     x  = (const float*)d_in[0];       // [100000,128]
    const float*     W1 = (const float*)d_in[1];       // [128,64]
    const float*     b1 = (const float*)d_in[2];       // [64]
    const float*     W2 = (const float*)d_in[3];       // [64,2]
    const float*     b2 = (const float*)d_in[4];       // [2]
    const long long* ei = (const long long*)d_in[5];   // [2,1200000] int64
    float*           out = (float*)d_out;              // [100000,2]

    // workspace layout (floats): H0 | H1 | dinv | O0 | O2  (~53 MB total)
    float* ws   = (float*)d_ws;
    float* H0   = ws;
    float* H1   = H0   + (size_t)N_NODES * HIDDEN;
    float* dinv = H1   + (size_t)N_NODES * HIDDEN;
    float* O0   = dinv + (size_t)N_NODES;
    float* O2   = O0   + (size_t)N_NODES * 2;

    const int T = 256;
    int grid_init  = (N_NODES * HIDDEN + T - 1) / T;       // 25000
    int grid_deg   = (N_EDGES + T - 1) / T;                // 4688
    int grid_node  = (N_NODES + T - 1) / T;                // 391
    int grid_gemm  = (N_WAVES * 32 + T - 1) / T;           // 782
    int grid_agg1  = (int)(((long)TOT_E * 32 + T - 1) / T);// 162500
    int grid_agg2  = (TOT_E + T - 1) / T;                  // 5079

    gcn_init_kernel  <<<grid_init, T, 0, stream>>>(dinv, H1, O2);
    gcn_degree_kernel<<<grid_deg,  T, 0, stream>>>(ei, dinv);
    gcn_rsqrt_kernel <<<grid_node, T, 0, stream>>>(dinv);
    gcn_gemm1_kernel <<<grid_gemm, T, 0, stream>>>(x, W1, H0);
    gcn_agg1_kernel  <<<grid_agg1, T, 0, stream>>>(ei, dinv, H0, H1);
    gcn_gemm2_kernel <<<grid_gemm, T, 0, stream>>>(H1, b1, W2, O0);
    gcn_agg2_kernel  <<<grid_agg2, T, 0, stream>>>(ei, dinv, O0, O2);
    gcn_final_kernel <<<grid_node, T, 0, stream>>>(O2, b2, out);
}